// GraphPool_39436389712229
// MI455X (gfx1250) — compile-verified
//
#include <hip/hip_runtime.h>
#include <hip/hip_bf16.h>

typedef float v2f __attribute__((ext_vector_type(2)));
typedef float v4f __attribute__((ext_vector_type(4)));
typedef float v8f __attribute__((ext_vector_type(8)));

#define NN 8192
#define DD 512
#define KK 4096

// ---------------- Kernel 1: Hp[i] = X[i,:].w + w_bias + gat_bias (wave per row)
__global__ void k_hp(const float* __restrict__ X, const float* __restrict__ w,
                     const float* __restrict__ wb, const float* __restrict__ gb,
                     float* __restrict__ Hp) {
  int lane = threadIdx.x & 31;
  int row  = (blockIdx.x * blockDim.x + threadIdx.x) >> 5;
  if (row >= NN) return;
  const float* xr = X + (size_t)row * DD;
  float acc = 0.f;
  for (int d = lane; d < DD; d += 32) acc = fmaf(xr[d], w[d], acc);
  for (int off = 16; off > 0; off >>= 1) acc += __shfl_xor(acc, off, 32);
  if (lane == 0) Hp[row] = acc + wb[0] + gb[0];
}

// ---------------- Kernel 2: global max of sa=phi0*Hp, then E=exp(sa-m), EHp=E*Hp (1 block)
__global__ void k_e(const float* __restrict__ Hp, const float* __restrict__ phi,
                    float* __restrict__ E, float* __restrict__ EHp) {
  __shared__ float red[32];
  const float phi0 = phi[0];
  int tid = threadIdx.x;
  float m = -1e30f;
  for (int i = tid; i < NN; i += blockDim.x) m = fmaxf(m, Hp[i] * phi0);
  for (int off = 16; off > 0; off >>= 1) m = fmaxf(m, __shfl_xor(m, off, 32));
  if ((tid & 31) == 0) red[tid >> 5] = m;
  __syncthreads();
  if (tid < 32) {
    float v = red[tid];
    for (int off = 16; off > 0; off >>= 1) v = fmaxf(v, __shfl_xor(v, off, 32));
    red[tid] = v;
  }
  __syncthreads();
  m = red[0];
  for (int i = tid; i < NN; i += blockDim.x) {
    float hp = Hp[i];
    float e  = __expf(hp * phi0 - m);
    E[i] = e;
    EHp[i] = e * hp;
  }
}

// ---------------- Kernel 3: masked-softmax GEMV over A via WMMA (branch-free hot loop).
// One wave per 16-row band. Per 8 columns: one b128 A load + one b128 B load per lane,
// feeding two V_WMMA_F32_16X16X4_F32. B col0=E, col1=E*Hp, cols 2..15 zeroed by bmask.
// den[i] = sum_j A[i,j]*E[j], num[i] = sum_j A[i,j]*E[j]*Hp[j].
__global__ void k_rows_wmma(const float* __restrict__ A, const float* __restrict__ E,
                            const float* __restrict__ EHp,
                            float* __restrict__ den, float* __restrict__ num) {
  const int lane = threadIdx.x & 31;
  const int wave = threadIdx.x >> 5;
  const int band = blockIdx.x * (blockDim.x >> 5) + wave;   // 16-row band id
  const int row0 = band * 16;
  const int n    = lane & 15;   // A: row-in-band; B/D: column
  const int half = lane >> 4;   // lo half: K={k,k+1}; hi half: K={k+4,k+5} (permuted K)
  const float* aRow  = A + (size_t)(row0 + n) * NN;
  const float* bbase = (n == 1) ? EHp : E;        // per-lane, loop-invariant
  const float  bmask = (n < 2) ? 1.0f : 0.0f;     // zero out B columns 2..15

  v8f c = {};
  for (int kt = 0; kt < NN; kt += 512) {
    __builtin_prefetch(aRow + kt + 2048, 0, 3);   // 8KB ahead, near caches
#pragma unroll 4
    for (int k0 = kt; k0 < kt + 512; k0 += 8) {
      const int kk = k0 + 4 * half;               // 16B aligned
      v4f a4 = *(const v4f*)(aRow + kk);
      v4f b4 = *(const v4f*)(bbase + kk);
      b4 *= bmask;
      v2f a0 = {a4.x, a4.y}, a1 = {a4.z, a4.w};
      v2f b0 = {b4.x, b4.y}, b1 = {b4.z, b4.w};
      c = __builtin_amdgcn_wmma_f32_16x16x4_f32(false, a0, false, b0, (short)0, c,
                                                false, false);
      c = __builtin_amdgcn_wmma_f32_16x16x4_f32(false, a1, false, b1, (short)0, c,
                                                false, false);
    }
  }
  // D 16x16 f32: vgpr r -> M=r (lanes 0-15) / M=r+8 (lanes 16-31), N=lane&15
  if (n < 2) {
    float* dst = (n == 0) ? den : num;
    const int mBase = row0 + half * 8;
#pragma unroll
    for (int r = 0; r < 8; ++r) dst[mBase + r] = c[r];
  }
}

// ---------------- Kernel 4: force diagonal weight to 1, h=num/den, score=sigmoid(h/100)
__global__ void k_finalize(const float* __restrict__ A, const float* __restrict__ E,
                           const float* __restrict__ EHp, const float* __restrict__ den,
                           const float* __restrict__ num, float* __restrict__ score) {
  int i = blockIdx.x * blockDim.x + threadIdx.x;
  if (i >= NN) return;
  float corr = 1.0f - A[(size_t)i * NN + i];   // mask = A + I  => diag weight == 1
  float d  = den[i] + corr * E[i];
  float nm = num[i] + corr * EHp[i];
  float h  = nm / d;
  score[i] = 1.0f / (1.0f + __expf(-h * 0.01f));
}

// ---------------- Kernel 5: single-block bitonic sort of 8192 (score desc, idx asc)
// 64-bit key = (~monotone_u32(score) << 32) | idx, sorted ascending.
__global__ __launch_bounds__(1024) void k_sort(const float* __restrict__ score,
                                               int* __restrict__ idx_out,
                                               float* __restrict__ val_out,
                                               float* __restrict__ out_idx_f) {
  __shared__ unsigned long long s[NN];   // 64 KB of the 320 KB WGP LDS
  const int tid = threadIdx.x;
  for (int i = tid; i < NN; i += 1024) {
    unsigned u = __float_as_uint(score[i]);
    u = (u & 0x80000000u) ? ~u : (u | 0x80000000u);           // ascending float map
    s[i] = ((unsigned long long)(~u) << 32) | (unsigned)i;    // desc score, asc idx
  }
  __syncthreads();
  for (int k = 2; k <= NN; k <<= 1) {
    for (int j = k >> 1; j > 0; j >>= 1) {
      for (int p = tid; p < NN / 2; p += 1024) {
        int i   = ((p & ~(j - 1)) << 1) | (p & (j - 1));
        int ixj = i + j;
        bool up = ((i & k) == 0);
        unsigned long long va = s[i], vb = s[ixj];
        if ((va > vb) == up) { s[i] = vb; s[ixj] = va; }
      }
      __syncthreads();
    }
  }
  for (int r = tid; r < KK; r += 1024) {
    int id = (int)(s[r] & 0xFFFFFFFFull);
    idx_out[r]   = id;
    val_out[r]   = score[id];
    out_idx_f[r] = (float)id;
  }
}

// ---------------- Kernel 6: A_new[r,c] = A[idx[r], idx[c]]  (idx cached in LDS)
__global__ void k_gather_A(const float* __restrict__ A, const int* __restrict__ idx,
                           float* __restrict__ outA) {
  __shared__ int sidx[KK];   // 16 KB
  for (int c = threadIdx.x; c < KK; c += blockDim.x) sidx[c] = idx[c];
  __syncthreads();
  const int r = blockIdx.x;
  const float* ar = A + (size_t)sidx[r] * NN;
  float* o = outA + (size_t)r * KK;
  for (int c = threadIdx.x; c < KK; c += blockDim.x) o[c] = ar[sidx[c]];
}

// ---------------- Kernel 7: new_X[r,:] = X[idx[r],:] * values[r]
__global__ void k_gather_X(const float* __restrict__ X, const int* __restrict__ idx,
                           const float* __restrict__ val, float* __restrict__ outX) {
  const int r = blockIdx.x;
  const int id = idx[r];
  const float v = val[r];
  const float* xr = X + (size_t)id * DD;
  float* o = outX + (size_t)r * DD;
  for (int d = threadIdx.x; d < DD; d += blockDim.x) o[d] = xr[d] * v;
}

extern "C" void kernel_launch(void* const* d_in, const int* in_sizes, int n_in,
                              void* d_out, int out_size, void* d_ws, size_t ws_size,
                              hipStream_t stream) {
  const float* A   = (const float*)d_in[0];   // [8192,8192]
  const float* X   = (const float*)d_in[1];   // [8192,512]
  const float* w   = (const float*)d_in[2];   // [1,512]
  const float* wb  = (const float*)d_in[3];   // [1]
  const float* gb  = (const float*)d_in[4];   // [1]
  const float* phi = (const float*)d_in[5];   // [2,1]

  float* ws  = (float*)d_ws;
  float* Hp  = ws;                 // 8192
  float* E   = ws + 1 * NN;        // 8192
  float* EHp = ws + 2 * NN;        // 8192
  float* den = ws + 3 * NN;        // 8192
  float* num = ws + 4 * NN;        // 8192
  float* sc  = ws + 5 * NN;        // 8192
  float* val = ws + 6 * NN;        // 4096 used
  int*   idx = (int*)(ws + 7 * NN);// 4096 used

  float* outA   = (float*)d_out;                         // 4096*4096
  float* outX   = outA + (size_t)KK * KK;                // 4096*512
  float* outIdx = outX + (size_t)KK * DD;                // 4096

  // 1) Hp: 8192 waves (wave per row)
  k_hp<<<NN * 32 / 256, 256, 0, stream>>>(X, w, wb, gb, Hp);
  // 2) global max + E / E*Hp
  k_e<<<1, 1024, 0, stream>>>(Hp, phi, E, EHp);
  // 3) WMMA masked row sums over A: 512 bands, 8 waves/block
  k_rows_wmma<<<(NN / 16) / 8, 256, 0, stream>>>(A, E, EHp, den, num);
  // 4) diagonal fix + sigmoid
  k_finalize<<<NN / 256, 256, 0, stream>>>(A, E, EHp, den, num, sc);
  // 5) full descending sort (top-k with jax tie-breaking), writes idx/values + idx output
  k_sort<<<1, 1024, 0, stream>>>(sc, idx, val, outIdx);
  // 6) A_new gather
  k_gather_A<<<KK, 256, 0, stream>>>(A, idx, outA);
  // 7) new_X gather+scale
  k_gather_X<<<KK, 256, 0, stream>>>(X, idx, val, outX);
}